// ComboLossHD_10505490006528
// MI455X (gfx1250) — compile-verified
//
#include <hip/hip_runtime.h>
#include <hip/hip_bf16.h>
#include <stdint.h>

#define ALPHA  0.8f
#define EDGE_W 1.0f
#define BOUND_W 1.0f
#define CE_W   1.0f
#define POS_W  1.0f

#define BATCH 16
#define HH 256
#define WW 256
#define NTOT (BATCH * HH * WW)
#define NBLK 4096              // B*H for pass2, B*(H/16)*(W/16) for edge pass
#define INF_F 131072.0f        // H*H + W*W, matches reference

typedef float v2f  __attribute__((ext_vector_type(2)));
typedef float v8f  __attribute__((ext_vector_type(8)));
typedef unsigned int u32x4 __attribute__((ext_vector_type(4)));
typedef int  i32x4 __attribute__((ext_vector_type(4)));
typedef int  i32x8 __attribute__((ext_vector_type(8)));

#if defined(__gfx1250__) && __has_builtin(__builtin_amdgcn_tensor_load_to_lds) && __has_builtin(__builtin_amdgcn_s_wait_tensorcnt)
#define HAVE_TDM 1
#else
#define HAVE_TDM 0
#endif

// ---------------- fast sigmoid (single v_rcp_f32 instead of IEEE divide) ----
__device__ inline float fast_sigmoid(float x) {
#if defined(__gfx1250__) && __has_builtin(__builtin_amdgcn_rcpf)
  return __builtin_amdgcn_rcpf(1.0f + expf(-x));
#else
  return 1.0f / (1.0f + expf(-x));
#endif
}

// ---------------- wave/block reductions (WMMA-accelerated cross-lane sum) ---
__device__ inline float wave_sum32(float v) {
#if defined(__gfx1250__) && __has_builtin(__builtin_amdgcn_wmma_f32_16x16x4_f32)
  // A(16x4): VGPR0 = lane value (K=0 / K=2 halves), VGPR1 = 0 (K=1 / K=3).
  // B(4x16) = all ones.  D[m][n] = val(m) + val(m+16) for every n.
  v2f a; a.x = v;    a.y = 0.0f;
  v2f b; b.x = 1.0f; b.y = 1.0f;
  v8f c = {};
  c = __builtin_amdgcn_wmma_f32_16x16x4_f32(false, a, false, b, (short)0, c,
                                            false, false);
  // lanes 0-15 hold rows 0..7 in c[0..7]; lanes 16-31 hold rows 8..15.
  float s = c[0] + c[1] + c[2] + c[3] + c[4] + c[5] + c[6] + c[7];
  s += __shfl_xor(s, 16, 32);
  return s;
#else
  for (int off = 16; off > 0; off >>= 1) v += __shfl_xor(v, off, 32);
  return v;
#endif
}

// blockDim.x must be 256 (8 waves)
__device__ inline float block_sum(float v, float* red8) {
  int tid = threadIdx.x;
  float w = wave_sum32(v);
  __syncthreads();                       // protect red8 across repeated calls
  if ((tid & 31) == 0) red8[tid >> 5] = w;
  __syncthreads();
  float r = 0.0f;
#pragma unroll
  for (int i = 0; i < 8; ++i) r += red8[i];
  return r;
}

// ---------------- TDM: DMA a 2D f32 tile (nrow rows of dim0 elems) to LDS ---
__device__ inline void tdm_load_2d_f32(const float* gsrc, void* lds_dst,
                                       int dim0, int nrow, int stride_elems) {
#if HAVE_TDM
  uint64_t ga = (uint64_t)(uintptr_t)gsrc;
  uint32_t la = (uint32_t)(uintptr_t)lds_dst;
  u32x4 g0;
  g0[0] = 1u;                                          // count=1, user mode
  g0[1] = la;                                          // lds_addr
  g0[2] = (uint32_t)(ga & 0xFFFFFFFFu);                // global_addr[31:0]
  g0[3] = (uint32_t)((ga >> 32) & 0x01FFFFFFu) | (2u << 30);  // addr[56:32]|type=2
  i32x8 g1;
  g1[0] = (2 << 16);                                   // data_size=4B
  g1[1] = (int)((unsigned)(dim0 & 0xFFFF) << 16);      // tensor_dim0[15:0]
  g1[2] = (int)(((unsigned)dim0 >> 16) & 0xFFFFu) |
          (int)((unsigned)(nrow & 0xFFFF) << 16);      // dim0 hi | tensor_dim1 lo
  g1[3] = (int)((unsigned)(dim0 & 0xFFFF) << 16);      // dim1 hi=0 | tile_dim0
  g1[4] = nrow;                                        // tile_dim1 | tile_dim2=0
  g1[5] = stride_elems;                                // tensor_dim0_stride lo32
  g1[6] = 0;                                           // stride0 hi | stride1 lo
  g1[7] = 0;
  i32x4 z4 = {0, 0, 0, 0};
#if __has_include(<hip/amd_detail/amd_gfx1250_TDM.h>)
  i32x8 z8 = {0, 0, 0, 0, 0, 0, 0, 0};
  __builtin_amdgcn_tensor_load_to_lds(g0, g1, z4, z4, z8, 0);
#else
  __builtin_amdgcn_tensor_load_to_lds(g0, g1, z4, z4, 0);
#endif
#else
  (void)gsrc; (void)lds_dst; (void)dim0; (void)nrow; (void)stride_elems;
#endif
}

// ---------------- init: zero has_pos flags ----------------------------------
__global__ void init_kernel(int* has_pos) {
  int tid = threadIdx.x;
  if (tid < BATCH) has_pos[tid] = 0;
}

// ---------------- pass 1: exact per-column 1D EDT (binary seeds) ------------
__global__ void edt_cols(const float* __restrict__ t,
                         float* __restrict__ gpos,
                         float* __restrict__ gneg,
                         int* __restrict__ has_pos) {
  int c = blockIdx.x * blockDim.x + threadIdx.x;   // 0 .. B*W-1
  if (c >= BATCH * WW) return;
  int b = c >> 8;
  int j = c & 255;
  const float* tb = t + (size_t)b * HH * WW + j;
  float* gp = gpos + (size_t)b * HH * WW + j;
  float* gn = gneg + (size_t)b * HH * WW + j;

  unsigned mask[8];
#pragma unroll
  for (int i = 0; i < 8; ++i) mask[i] = 0u;

  const int BIGI = 1 << 20;
  int lastP = -BIGI, lastN = -BIGI;
  bool anyP = false;
  for (int xr = 0; xr < HH; ++xr) {
    float tv = tb[(size_t)xr * WW];
    bool pos = tv > 0.5f;
    if (pos) { mask[xr >> 5] |= (1u << (xr & 31)); lastP = xr; anyP = true; }
    else     { lastN = xr; }
    gp[(size_t)xr * WW] = (float)(xr - lastP);     // forward distance
    gn[(size_t)xr * WW] = (float)(xr - lastN);
  }
  if (anyP) atomicOr(&has_pos[b], 1);

  int nextP = 3 * BIGI, nextN = 3 * BIGI;
  for (int xr = HH - 1; xr >= 0; --xr) {
    bool pos = (mask[xr >> 5] >> (xr & 31)) & 1u;
    if (pos) nextP = xr; else nextN = xr;
    float dP = fminf(gp[(size_t)xr * WW], (float)(nextP - xr));
    float dN = fminf(gn[(size_t)xr * WW], (float)(nextN - xr));
    gp[(size_t)xr * WW] = (dP >= (float)BIGI) ? INF_F : dP * dP;
    gn[(size_t)xr * WW] = (dN >= (float)BIGI) ? INF_F : dN * dN;
  }
}

// ---------------- pass 2: row min-plus out of LDS + fused boundary ----------
__global__ void edt_pass2_boundary(const float* __restrict__ x,
                                   const float* __restrict__ t,
                                   const float* __restrict__ gpos,
                                   const float* __restrict__ gneg,
                                   const int* __restrict__ has_pos,
                                   float* __restrict__ partialB) {
  __shared__ float rows[2 * WW];     // [0..255] = gpos row, [256..511] = gneg
  __shared__ float red[8];
  int bi = blockIdx.x;               // enumerates (b, row) = b*H + xr
  int b = bi >> 8;
  int tid = threadIdx.x;
  const float* gp = gpos + (size_t)bi * WW;

#if HAVE_TDM
  if (tid < 32) {                    // wave 0 issues one 2D DMA descriptor:
    // row 0 = gpos row, row 1 = gneg row (gneg = gpos + NTOT elements)
    tdm_load_2d_f32(gp, rows, WW, 2, NTOT);
    __builtin_amdgcn_s_wait_tensorcnt(0);
  }
  __syncthreads();
#else
  rows[tid] = gp[tid];
  rows[WW + tid] = gp[NTOT + tid];
  __syncthreads();
#endif

  int y = tid;
  float dp = 1e30f, dn = 1e30f;
#pragma unroll 4
  for (int j = 0; j < WW; ++j) {
    float dy = (float)(y - j);
    float q = dy * dy;
    dp = fminf(dp, rows[j] + q);
    dn = fminf(dn, rows[WW + j] + q);
  }
  // d_out = dist with seed=~pos (gneg); d_in = seed=pos (gpos)
  float sdf = (has_pos[b] != 0) ? (sqrtf(dn) - sqrtf(dp)) : 0.0f;
  size_t idx = (size_t)bi * WW + y;
  float s = fast_sigmoid(x[idx]);
  float be = (s - t[idx]) * sdf;
  float bs = block_sum(be, red);
  if (tid == 0) partialB[bi] = bs;
}

// ---------------- pass 3: fused focal + CE + sobel edge ---------------------
__global__ void edge_focal_kernel(const float* __restrict__ x,
                                  const float* __restrict__ t,
                                  float* __restrict__ pF,
                                  float* __restrict__ pC,
                                  float* __restrict__ pE) {
  __shared__ float sS[18][18];     // sigmoid(x) tile with halo (replicate pad)
  __shared__ float sT[18][18];     // target tile with halo
  __shared__ float red[8];
  int tx = blockIdx.x * 16;
  int ty = blockIdx.y * 16;
  int b = blockIdx.z;
  const float* xb = x + (size_t)b * HH * WW;
  const float* tb = t + (size_t)b * HH * WW;
  int tid = threadIdx.x;

  for (int i = tid; i < 18 * 18; i += 256) {
    int r = i / 18, c = i % 18;
    int gr = min(max(ty + r - 1, 0), HH - 1);
    int gc = min(max(tx + c - 1, 0), WW - 1);
    float xv = xb[(size_t)gr * WW + gc];
    sS[r][c] = fast_sigmoid(xv);
    sT[r][c] = tb[(size_t)gr * WW + gc];
  }
  __syncthreads();

  int lr = tid >> 4, lc = tid & 15;
  int r = lr + 1, c = lc + 1;

  float a00 = sS[r-1][c-1], a01 = sS[r-1][c], a02 = sS[r-1][c+1];
  float a10 = sS[r  ][c-1],                   a12 = sS[r  ][c+1];
  float a20 = sS[r+1][c-1], a21 = sS[r+1][c], a22 = sS[r+1][c+1];
  float gxs = ((a02 + 2.0f * a12 + a22) - (a00 + 2.0f * a10 + a20)) * 0.125f;
  float gys = ((a20 + 2.0f * a21 + a22) - (a00 + 2.0f * a01 + a02)) * 0.125f;

  float b00 = sT[r-1][c-1], b01 = sT[r-1][c], b02 = sT[r-1][c+1];
  float b10 = sT[r  ][c-1],                   b12 = sT[r  ][c+1];
  float b20 = sT[r+1][c-1], b21 = sT[r+1][c], b22 = sT[r+1][c+1];
  float gxt = ((b02 + 2.0f * b12 + b22) - (b00 + 2.0f * b10 + b20)) * 0.125f;
  float gyt = ((b20 + 2.0f * b21 + b22) - (b00 + 2.0f * b01 + b02)) * 0.125f;

  float ms = sqrtf(gxs * gxs + gys * gys + 1e-6f);
  float mt = sqrtf(gxt * gxt + gyt * gyt + 1e-6f);
  float edge_e = fabsf(ms - mt);

  // focal / ce on center pixel (stable log-sigmoid)
  float xv = xb[(size_t)(ty + lr) * WW + (tx + lc)];
  float tv = sT[r][c];
  float lp = (xv >= 0.0f) ? -log1pf(expf(-xv)) : (xv - log1pf(expf(xv)));
  float ln = lp - xv;                       // log_sigmoid(-x)
  float bce = -(tv * lp + (1.0f - tv) * ln);
  float pt = expf(-bce);
  float om = 1.0f - pt;
  float fo = ALPHA * om * om * bce;         // GAMMA = 2
  float ce = -(POS_W * tv * lp + (1.0f - tv) * ln);

  float sF = block_sum(fo,     red);
  float sC = block_sum(ce,     red);
  float sE = block_sum(edge_e, red);
  if (tid == 0) {
    int pidx = (blockIdx.z * gridDim.y + blockIdx.y) * gridDim.x + blockIdx.x;
    pF[pidx] = sF;
    pC[pidx] = sC;
    pE[pidx] = sE;
  }
}

// ---------------- final deterministic combine -------------------------------
__global__ void combine_kernel(const float* __restrict__ pB,
                               const float* __restrict__ pF,
                               const float* __restrict__ pC,
                               const float* __restrict__ pE,
                               float* __restrict__ out) {
  __shared__ float red[8];
  int tid = threadIdx.x;
  float sB = 0.f, sF = 0.f, sC = 0.f, sE = 0.f;
  for (int i = tid; i < NBLK; i += 256) {
    sB += pB[i]; sF += pF[i]; sC += pC[i]; sE += pE[i];
  }
  sB = block_sum(sB, red);
  sF = block_sum(sF, red);
  sC = block_sum(sC, red);
  sE = block_sum(sE, red);
  if (tid == 0) {
    float invN = 1.0f / (float)NTOT;
    out[0] = (sF + EDGE_W * sE + BOUND_W * sB + CE_W * sC) * invN;
  }
}

extern "C" void kernel_launch(void* const* d_in, const int* in_sizes, int n_in,
                              void* d_out, int out_size, void* d_ws,
                              size_t ws_size, hipStream_t stream) {
  (void)in_sizes; (void)n_in; (void)out_size; (void)ws_size;
  const float* x = (const float*)d_in[0];
  const float* t = (const float*)d_in[1];
  float* ws   = (float*)d_ws;
  float* gpos = ws;                      // N floats
  float* gneg = gpos + NTOT;             // N floats (must stay adjacent: pass2
                                         // loads both rows with one 2D TDM)
  float* pB   = gneg + NTOT;             // 4096
  float* pF   = pB + NBLK;               // 4096
  float* pC   = pF + NBLK;               // 4096
  float* pE   = pC + NBLK;               // 4096
  int* has_pos = (int*)(pE + NBLK);      // 16 ints
  float* out = (float*)d_out;

  init_kernel<<<1, 64, 0, stream>>>(has_pos);
  edt_cols<<<(BATCH * WW) / 256, 256, 0, stream>>>(t, gpos, gneg, has_pos);
  edt_pass2_boundary<<<BATCH * HH, 256, 0, stream>>>(x, t, gpos, gneg, has_pos,
                                                     pB);
  edge_focal_kernel<<<dim3(WW / 16, HH / 16, BATCH), 256, 0, stream>>>(x, t, pF,
                                                                       pC, pE);
  combine_kernel<<<1, 256, 0, stream>>>(pB, pF, pC, pE, out);
}